// GAT_3023656976827
// MI455X (gfx1250) — compile-verified
//
#include <hip/hip_runtime.h>
#include <math.h>

#define NN   89250
#define NE   899756
#define ETOT (NE + NN)      // 989006 edges incl. self loops
#define FIN  500
#define H1   8
#define C1   16
#define F1   (H1 * C1)      // 128
#define C2   7
#define SLOPE 0.2f

typedef float v2f __attribute__((ext_vector_type(2)));
typedef float v8f __attribute__((ext_vector_type(8)));

__device__ inline float lrelu(float x) { return x > 0.f ? x : SLOPE * x; }

__device__ inline float att4(float4 w, float4 a, float4 r) {
    return w.x * lrelu(a.x + r.x) + w.y * lrelu(a.y + r.y) +
           w.z * lrelu(a.z + r.z) + w.w * lrelu(a.w + r.w);
}

__device__ inline float4 upd4(float4 acc, float s, float e, float4 a) {
    float4 o;
    o.x = acc.x * s + e * a.x;  o.y = acc.y * s + e * a.y;
    o.z = acc.z * s + e * a.z;  o.w = acc.w * s + e * a.w;
    return o;
}

// ---------------- CSR build ----------------

__global__ void fill_i32(int* __restrict__ p, int v, int n) {
    int i = blockIdx.x * blockDim.x + threadIdx.x;
    if (i < n) p[i] = v;
}

__global__ void count_deg(const int* __restrict__ ei, int* __restrict__ deg) {
    int e = blockIdx.x * blockDim.x + threadIdx.x;
    if (e >= ETOT) return;
    int d = (e < NE) ? ei[NE + e] : (e - NE);
    atomicAdd(&deg[d], 1);
}

// single-block exclusive scan over deg -> base (and cursor copy)
__global__ void __launch_bounds__(1024)
scan_excl(const int* __restrict__ deg, int* __restrict__ base, int* __restrict__ cursor) {
    __shared__ int tmp[1024];
    __shared__ int carry_s;
    const int tid = threadIdx.x;
    if (tid == 0) carry_s = 0;
    __syncthreads();
    for (int start = 0; start < NN; start += 1024) {
        int i = start + tid;
        int v = (i < NN) ? deg[i] : 0;
        tmp[tid] = v;
        __syncthreads();
        #pragma unroll
        for (int off = 1; off < 1024; off <<= 1) {
            int t = (tid >= off) ? tmp[tid - off] : 0;
            __syncthreads();
            tmp[tid] += t;
            __syncthreads();
        }
        int excl = carry_s + tmp[tid] - v;
        if (i < NN) { base[i] = excl; cursor[i] = excl; }
        __syncthreads();
        if (tid == 1023) carry_s += tmp[1023];
        __syncthreads();
    }
}

__global__ void scatter_edges(const int* __restrict__ ei, int* __restrict__ cursor,
                              int* __restrict__ adj) {
    int e = blockIdx.x * blockDim.x + threadIdx.x;
    if (e >= ETOT) return;
    int s = (e < NE) ? ei[e]      : (e - NE);
    int d = (e < NE) ? ei[NE + e] : (e - NE);
    int pos = atomicAdd(&cursor[d], 1);
    adj[pos] = s;
}

// ---------------- layer-1 GEMM via V_WMMA_F32_16X16X4_F32 ----------------
// One block: 16 waves. A-tile (16 x 500) staged in LDS once; waves 0-7 compute
// the 8 column tiles of X@W1l, waves 8-15 those of X@W1r.
__global__ void __launch_bounds__(512)
gemm1_wmma(const float* __restrict__ X,
           const float* __restrict__ Wl, const float* __restrict__ bl,
           const float* __restrict__ Wr, const float* __restrict__ br,
           float* __restrict__ xl, float* __restrict__ xr)
{
    __shared__ float Alds[16 * FIN];   // 32 KB
    const int tid = threadIdx.x;
    const int m0  = blockIdx.x * 16;

    for (int idx = tid; idx < 16 * FIN; idx += 512) {
        int r = idx / FIN, c = idx - r * FIN;
        int row = m0 + r; if (row >= NN) row = NN - 1;     // clamp; masked on store
        Alds[idx] = X[(size_t)row * FIN + c];
    }
    __syncthreads();

    const int lane = tid & 31;
    const int wave = tid >> 5;
    const int yy   = wave >> 3;         // 0 -> Wl, 1 -> Wr
    const int n0   = (wave & 7) * 16;

    const float* __restrict__ W    = yy ? Wr : Wl;
    const float* __restrict__ bias = yy ? br : bl;
    float*       __restrict__ out  = yy ? xr : xl;

    const int half = lane >> 4;         // 0: K={0,1}, 1: K={2,3}
    const int l15  = lane & 15;
    const float* __restrict__ Ap = &Alds[l15 * FIN + 2 * half];
    const float* __restrict__ Bp = W + (size_t)(2 * half) * F1 + n0 + l15;

    v8f acc0 = {0.f,0.f,0.f,0.f,0.f,0.f,0.f,0.f};
    v8f acc1 = {0.f,0.f,0.f,0.f,0.f,0.f,0.f,0.f};

    int k = 0;
    for (; k + 8 <= FIN; k += 8) {
        v2f a0; a0.x = Ap[k];     a0.y = Ap[k + 1];
        v2f b0; b0.x = Bp[(size_t)k * F1];       b0.y = Bp[(size_t)(k + 1) * F1];
        acc0 = __builtin_amdgcn_wmma_f32_16x16x4_f32(false, a0, false, b0,
                                                     (short)0, acc0, false, false);
        v2f a1; a1.x = Ap[k + 4]; a1.y = Ap[k + 5];
        v2f b1; b1.x = Bp[(size_t)(k + 4) * F1]; b1.y = Bp[(size_t)(k + 5) * F1];
        acc1 = __builtin_amdgcn_wmma_f32_16x16x4_f32(false, a1, false, b1,
                                                     (short)0, acc1, false, false);
    }
    for (; k < FIN; k += 4) {
        v2f a; a.x = Ap[k]; a.y = Ap[k + 1];
        v2f b; b.x = Bp[(size_t)k * F1]; b.y = Bp[(size_t)(k + 1) * F1];
        acc0 = __builtin_amdgcn_wmma_f32_16x16x4_f32(false, a, false, b,
                                                     (short)0, acc0, false, false);
    }
    v8f acc = acc0 + acc1;

    const int col = n0 + l15;
    const float bv = bias[col];
    #pragma unroll
    for (int r = 0; r < 8; ++r) {
        int row = m0 + half * 8 + r;
        if (row < NN) out[(size_t)row * F1 + col] = acc[r] + bv;
    }
}

// ---------------- layer-1 fused attention: online softmax per (node, head) ----
// Reads xl (source transforms) and xr (target transforms); writes ELU'd layer-1
// output in place over xr (each thread touches only its own 16-channel slice).
__global__ void gat_node1(const int* __restrict__ base, const int* __restrict__ deg,
                          const int* __restrict__ adj,
                          const float* __restrict__ xl, float* __restrict__ xr_h,
                          const float* __restrict__ att, const float* __restrict__ bias)
{
    int t = blockIdx.x * blockDim.x + threadIdx.x;
    if (t >= NN * H1) return;
    int n = t >> 3, h = t & 7;

    const float4* __restrict__ rp = (const float4*)(xr_h + (size_t)n * F1 + h * C1);
    float4 r0 = rp[0], r1 = rp[1], r2 = rp[2], r3 = rp[3];
    const float4* __restrict__ wp = (const float4*)(att + h * C1);
    float4 w0 = wp[0], w1 = wp[1], w2 = wp[2], w3 = wp[3];

    float m = -INFINITY, dsum = 0.f;
    float4 acc0 = {0,0,0,0}, acc1 = {0,0,0,0}, acc2 = {0,0,0,0}, acc3 = {0,0,0,0};

    const int b0 = base[n], dg = deg[n];
    for (int i = 0; i < dg; ++i) {
        int s = adj[b0 + i];
        const float4* __restrict__ pl = (const float4*)(xl + (size_t)s * F1 + h * C1);
        float4 a0 = pl[0], a1 = pl[1], a2 = pl[2], a3 = pl[3];
        float sc = att4(w0, a0, r0) + att4(w1, a1, r1) +
                   att4(w2, a2, r2) + att4(w3, a3, r3);
        float nm    = fmaxf(m, sc);
        float scale = __expf(m - nm);     // 0 on first edge (m = -inf)
        float ex    = __expf(sc - nm);
        dsum = dsum * scale + ex;
        acc0 = upd4(acc0, scale, ex, a0);
        acc1 = upd4(acc1, scale, ex, a1);
        acc2 = upd4(acc2, scale, ex, a2);
        acc3 = upd4(acc3, scale, ex, a3);
        m = nm;
    }
    float inv = 1.f / dsum;               // dg >= 1 (self loop)
    float* __restrict__ op = xr_h + (size_t)n * F1 + h * C1;
    const float* __restrict__ bp = bias + h * C1;
    float v[16] = { acc0.x, acc0.y, acc0.z, acc0.w, acc1.x, acc1.y, acc1.z, acc1.w,
                    acc2.x, acc2.y, acc2.z, acc2.w, acc3.x, acc3.y, acc3.z, acc3.w };
    #pragma unroll
    for (int c = 0; c < 16; ++c) {
        float o = v[c] * inv + bp[c];
        op[c] = o > 0.f ? o : (__expf(o) - 1.f);   // ELU
    }
}

// ---------------- layer-2 small GEMM (128 -> 7, both Wl and Wr) ----------------
__global__ void gemm2(const float* __restrict__ h,
                      const float* __restrict__ W2l, const float* __restrict__ b2l,
                      const float* __restrict__ W2r, const float* __restrict__ b2r,
                      float* __restrict__ xl2, float* __restrict__ xr2)
{
    int t = blockIdx.x * blockDim.x + threadIdx.x;
    if (t >= NN * C2) return;
    int n = t / C2, c = t % C2;
    const float* __restrict__ hp = h + (size_t)n * F1;
    float sl = 0.f, sr = 0.f;
    #pragma unroll 8
    for (int k = 0; k < F1; ++k) {
        float hv = hp[k];
        sl += hv * W2l[k * C2 + c];
        sr += hv * W2r[k * C2 + c];
    }
    xl2[t] = sl + b2l[c];
    xr2[t] = sr + b2r[c];
}

// ---------------- layer-2 fused attention (1 head, 7 channels) ----------------
__global__ void gat_node2(const int* __restrict__ base, const int* __restrict__ deg,
                          const int* __restrict__ adj,
                          const float* __restrict__ xl2, const float* __restrict__ xr2,
                          const float* __restrict__ att, const float* __restrict__ bias,
                          float* __restrict__ out)
{
    int n = blockIdx.x * blockDim.x + threadIdx.x;
    if (n >= NN) return;

    float r[C2], w[C2], acc[C2];
    #pragma unroll
    for (int c = 0; c < C2; ++c) { r[c] = xr2[(size_t)n * C2 + c]; w[c] = att[c]; acc[c] = 0.f; }

    float m = -INFINITY, dsum = 0.f;
    const int b0 = base[n], dg = deg[n];
    for (int i = 0; i < dg; ++i) {
        int s = adj[b0 + i];
        const float* __restrict__ pl = xl2 + (size_t)s * C2;
        float a[C2], sc = 0.f;
        #pragma unroll
        for (int c = 0; c < C2; ++c) { a[c] = pl[c]; sc += w[c] * lrelu(a[c] + r[c]); }
        float nm    = fmaxf(m, sc);
        float scale = __expf(m - nm);
        float ex    = __expf(sc - nm);
        dsum = dsum * scale + ex;
        #pragma unroll
        for (int c = 0; c < C2; ++c) acc[c] = acc[c] * scale + ex * a[c];
        m = nm;
    }
    float inv = 1.f / dsum;
    #pragma unroll
    for (int c = 0; c < C2; ++c) out[(size_t)n * C2 + c] = acc[c] * inv + bias[c];
}

// ---------------- host launcher ----------------
extern "C" void kernel_launch(void* const* d_in, const int* in_sizes, int n_in,
                              void* d_out, int out_size, void* d_ws, size_t ws_size,
                              hipStream_t stream) {
    const float* x     = (const float*)d_in[0];
    const int*   ei    = (const int*)  d_in[1];
    const float* W1l   = (const float*)d_in[2];
    const float* b1l   = (const float*)d_in[3];
    const float* W1r   = (const float*)d_in[4];
    const float* b1r   = (const float*)d_in[5];
    const float* att1  = (const float*)d_in[6];
    const float* bias1 = (const float*)d_in[7];
    const float* W2l   = (const float*)d_in[8];
    const float* b2l   = (const float*)d_in[9];
    const float* W2r   = (const float*)d_in[10];
    const float* b2r   = (const float*)d_in[11];
    const float* att2  = (const float*)d_in[12];
    const float* bias2 = (const float*)d_in[13];
    float* out = (float*)d_out;

    float* ws   = (float*)d_ws;
    float* xl1  = ws;                           // 128*N
    float* xr1  = xl1 + (size_t)F1 * NN;        // 128*N  (layer-1 output in place)
    float* xl2  = xr1 + (size_t)F1 * NN;        // 7*N
    float* xr2  = xl2 + (size_t)C2 * NN;        // 7*N
    int*   deg    = (int*)(xr2 + (size_t)C2 * NN);  // N
    int*   basep  = deg + NN;                   // N
    int*   cursor = basep + NN;                 // N
    int*   adj    = cursor + NN;                // ETOT

    const int B = 256;

    // ---- CSR by destination (built once, used by both layers) ----
    fill_i32<<<(NN + B - 1) / B, B, 0, stream>>>(deg, 0, NN);
    count_deg<<<(ETOT + B - 1) / B, B, 0, stream>>>(ei, deg);
    scan_excl<<<1, 1024, 0, stream>>>(deg, basep, cursor);
    scatter_edges<<<(ETOT + B - 1) / B, B, 0, stream>>>(ei, cursor, adj);

    // ---- layer 1 ----
    gemm1_wmma<<<(NN + 15) / 16, 512, 0, stream>>>(x, W1l, b1l, W1r, b1r, xl1, xr1);
    gat_node1<<<(NN * H1 + B - 1) / B, B, 0, stream>>>(basep, deg, adj, xl1, xr1, att1, bias1);

    // ---- layer 2 ----
    gemm2<<<(NN * C2 + B - 1) / B, B, 0, stream>>>(xr1, W2l, b2l, W2r, b2r, xl2, xr2);
    gat_node2<<<(NN + B - 1) / B, B, 0, stream>>>(basep, deg, adj, xl2, xr2, att2, bias2, out);
}